// DenseImageWarp_30812095382185
// MI455X (gfx1250) — compile-verified
//
#include <hip/hip_runtime.h>

#ifndef __has_builtin
#define __has_builtin(x) 0
#endif

#if __has_builtin(__builtin_amdgcn_global_load_async_to_lds_b32)
#define USE_ASYNC_LDS 1
#else
#define USE_ASYNC_LDS 0
#endif

namespace {

constexpr int B = 4, C = 64, H = 512, W = 512;
constexpr int HW = H * W;
constexpr int BLOCK = 256;

typedef __attribute__((address_space(1))) int g1_int;  // global
typedef __attribute__((address_space(3))) int l3_int;  // LDS
typedef float v2f __attribute__((ext_vector_type(2)));

// 8-byte load with only 4-byte alignment guarantee (gather column offsets are
// arbitrary). memcpy lets clang emit global_load_b64 when unaligned vmem is
// allowed, or split safely otherwise.
__device__ inline v2f load_pair(const float* p) {
  v2f v;
  __builtin_memcpy(&v, p, sizeof(v));
  return v;
}

__global__ __launch_bounds__(BLOCK) void
dense_image_warp_kernel(const float* __restrict__ image,
                        const float* __restrict__ flow,
                        float* __restrict__ out) {
  __shared__ float s_dy[BLOCK];
  __shared__ float s_dx[BLOCK];

  const int t   = threadIdx.x;
  const int pix = blockIdx.x * BLOCK + t;   // 0 .. B*H*W-1 (grid divides exactly)
  const int x   = pix & (W - 1);
  const int y   = (pix / W) & (H - 1);
  const int b   = pix / (H * W);

  // flow[b,0,y,x] = dy ; flow[b,1,y,x] = dx
  const size_t f_dy = (size_t)b * 2 * HW + (size_t)y * W + x;
  const size_t f_dx = f_dy + HW;

#if USE_ASYNC_LDS
  // CDNA5 async copy path: global -> LDS, tracked by ASYNCcnt.
  __builtin_amdgcn_global_load_async_to_lds_b32(
      (g1_int*)(flow + f_dy), (l3_int*)(&s_dy[t]), 0, 0);
  __builtin_amdgcn_global_load_async_to_lds_b32(
      (g1_int*)(flow + f_dx), (l3_int*)(&s_dx[t]), 0, 0);
#if __has_builtin(__builtin_amdgcn_s_wait_asynccnt)
  __builtin_amdgcn_s_wait_asynccnt(0);
#else
  asm volatile("s_wait_asynccnt 0" ::: "memory");
#endif
  __syncthreads();
#else
  s_dy[t] = flow[f_dy];
  s_dx[t] = flow[f_dx];
  __syncthreads();
#endif

  const float dy = s_dy[t];
  const float dx = s_dx[t];

  // Source coordinates: reference's normalize/unnormalize round trip, then
  // border clamp.
  float yf = (float)y - dy;
  float xf = (float)x - dx;
  const float sy = (float)(H - 1) * 0.5f;
  const float sx = (float)(W - 1) * 0.5f;
  const float gy = yf / sy - 1.0f;
  const float gx = xf / sx - 1.0f;
  yf = (gy + 1.0f) * 0.5f * (float)(H - 1);
  xf = (gx + 1.0f) * 0.5f * (float)(W - 1);
  xf = fminf(fmaxf(xf, 0.0f), (float)(W - 1));
  yf = fminf(fmaxf(yf, 0.0f), (float)(H - 1));

  const float x0f = floorf(xf);
  const float y0f = floorf(yf);
  float wx = xf - x0f;
  const float wy = yf - y0f;
  const int x0 = (int)x0f;
  const int y0 = (int)y0f;
  const int y1 = min(y0 + 1, H - 1);

  // Horizontal pair trick: x1 == x0+1 unless x0 == W-1 (then wx == 0).
  // Load the pair at xc = min(x0, W-2); if shifted, put all horizontal weight
  // on pair.y (== img[W-1]) by forcing wx = 1. pair.x then has weight exactly
  // 0, so the shifted value never contributes.
  const int xc = min(x0, W - 2);
  if (x0 == W - 1) wx = 1.0f;

  // Channel-invariant bilinear weights (computed once per pixel).
  const float u0 = 1.0f - wx, u1 = wx;
  const float v0 = 1.0f - wy, v1 = wy;
  const float w00 = v0 * u0, w01 = v0 * u1;
  const float w10 = v1 * u0, w11 = v1 * u1;

  // Per-channel gather bases; channel stride is HW floats.
  const float* imgb = image + (size_t)b * C * HW;
  const float* pT = imgb + (size_t)y0 * W + xc;  // top pair    (y0, xc..xc+1)
  const float* pBt = imgb + (size_t)y1 * W + xc; // bottom pair (y1, xc..xc+1)
  float* po = out + (size_t)b * C * HW + (size_t)y * W + x;

#pragma unroll 8
  for (int c = 0; c < C; ++c) {
    const size_t o = (size_t)c * HW;
    const v2f tp = load_pair(pT + o);
    const v2f bp = load_pair(pBt + o);
    const float r = fmaf(tp.x, w00,
                    fmaf(tp.y, w01,
                    fmaf(bp.x, w10, bp.y * w11)));
    // Output is streamed exactly once: non-temporal store keeps the 192MB L2
    // free for gather-neighborhood reuse.
    __builtin_nontemporal_store(r, po + o);
  }
}

}  // namespace

extern "C" void kernel_launch(void* const* d_in, const int* in_sizes, int n_in,
                              void* d_out, int out_size, void* d_ws, size_t ws_size,
                              hipStream_t stream) {
  (void)in_sizes; (void)n_in; (void)out_size; (void)d_ws; (void)ws_size;
  const float* image = (const float*)d_in[0];
  const float* flow  = (const float*)d_in[1];
  float* out = (float*)d_out;

  const int pixels = B * H * W;            // 1,048,576
  dense_image_warp_kernel<<<pixels / BLOCK, BLOCK, 0, stream>>>(image, flow, out);
}